// TripletLoss_3109556322825
// MI455X (gfx1250) — compile-verified
//
#include <hip/hip_runtime.h>
#include <hip/hip_bf16.h>
#include <math.h>

#define NROWS 8192
#define DIM   256
#define NTILE (NROWS / 16)   // 512 column tiles
#define KSTEPS (DIM / 4)     // 64 WMMA K-steps per tile
#define PIPE   16            // B-fragment pipeline depth (asm-enforced)
#define MARGIN_F 0.3f
#define NEG_FALLBACK_F 1.0e6f

typedef __attribute__((ext_vector_type(2))) float v2f;
typedef __attribute__((ext_vector_type(8))) float v8f;

// ---------------------------------------------------------------------------
// Kernel 0: squared row norms -> d_ws; also zero the scalar output.
// ---------------------------------------------------------------------------
__global__ __launch_bounds__(256) void sqnorm_kernel(const float* __restrict__ feat,
                                                     float* __restrict__ sqn,
                                                     float* __restrict__ out) {
    const int tid  = threadIdx.x;
    const int wave = tid >> 5;
    const int lane = tid & 31;
    const int row  = blockIdx.x * 8 + wave;

    if (blockIdx.x == 0 && tid == 0) out[0] = 0.0f;

    const float* rp = feat + (size_t)row * DIM;
    float acc = 0.0f;
#pragma unroll
    for (int i = 0; i < DIM / 32; ++i) {
        float v = rp[lane + 32 * i];
        acc = fmaf(v, v, acc);
    }
#pragma unroll
    for (int off = 16; off >= 1; off >>= 1)
        acc += __shfl_xor(acc, off, 32);
    if (lane == 0) sqn[row] = acc;
}

// ---------------------------------------------------------------------------
// Kernel 1: fused cdist + batch-hard mining + margin loss.
// Block = 256 threads = 8 waves, owns 16 rows. A-fragments live in registers.
// B-fragments flow through an asm-enforced 16-deep global_load_b64 pipeline:
// issue 16 loads, then per step "wait loadcnt<=15 -> wmma -> reissue slot".
// ---------------------------------------------------------------------------
__global__ __launch_bounds__(256) void triplet_hard_kernel(const float* __restrict__ feat,
                                                           const int*   __restrict__ labels,
                                                           const float* __restrict__ sqn,
                                                           float* __restrict__ out) {
    __shared__ float ldsA[16 * DIM];       // 16 KB: this block's 16 A-rows
    __shared__ float pAP[8][16];           // per-wave partial max(d_ap)
    __shared__ float pAN[8][16];           // per-wave partial min(d_an)

    const int tid   = threadIdx.x;
    const int wave  = tid >> 5;
    const int lane  = tid & 31;
    const int hl    = lane >> 4;           // lane half (0: lanes 0-15, 1: 16-31)
    const int ln    = lane & 15;
    const int mBase = blockIdx.x * 16;

    // Stage the 16 A-rows (row-major) into LDS.
    for (int i = tid; i < 16 * DIM; i += 256)
        ldsA[i] = feat[(size_t)mBase * DIM + i];
    __syncthreads();

    // Hoist all 64 A-fragments into registers (128 VGPRs), loaded once.
    const float* aptr = &ldsA[ln * DIM + hl * 2];
    v2f af[KSTEPS];
#pragma unroll
    for (int s = 0; s < KSTEPS; ++s)
        af[s] = *(const v2f*)(aptr + 4 * s);

    // Per-lane row metadata: accumulator VGPR r maps to M = r + 8*hl.
    int   lbl_m[8];
    float sqn_m[8];
#pragma unroll
    for (int r = 0; r < 8; ++r) {
        const int m = mBase + r + 8 * hl;
        lbl_m[r] = labels[m];
        sqn_m[r] = sqn[m];
    }

    float maxap[8], minan[8];
#pragma unroll
    for (int r = 0; r < 8; ++r) {
        maxap[r] = -__builtin_inff();
        minan[r] =  __builtin_inff();
    }

    // 512 column tiles round-robined over 8 waves -> 64 uniform iterations
    // (EXEC stays all-ones through every WMMA).
    for (int t = wave; t < NTILE; t += 8) {
        const int n      = t * 16 + ln;
        const int lbl_n  = labels[n];
        const float sq_n = sqn[n];
        const float* bptr = feat + (size_t)n * DIM + hl * 2;

        // ---- asm-enforced 16-deep B pipeline -------------------------------
        v2f bf[PIPE];
#pragma unroll
        for (int j = 0; j < PIPE; ++j)
            asm volatile("global_load_b64 %0, %1, off offset:%2"
                         : "=v"(bf[j])
                         : "v"(bptr), "i"(16 * j));

        v8f acc = {};
#pragma unroll
        for (int s = 0; s < KSTEPS; ++s) {
            // Guarantee load s has landed: <=15 newer loads outstanding.
            asm volatile("s_wait_loadcnt 0xf" : "+v"(bf[s & (PIPE - 1)]));
            acc = __builtin_amdgcn_wmma_f32_16x16x4_f32(
                false, af[s], false, bf[s & (PIPE - 1)], (short)0, acc,
                false, false);
            if (s + PIPE < KSTEPS)
                asm volatile("global_load_b64 %0, %1, off offset:%2"
                             : "=v"(bf[s & (PIPE - 1)])
                             : "v"(bptr), "i"(16 * (s + PIPE)));
        }
        // --------------------------------------------------------------------

        // dist tile -> masked hard mining
#pragma unroll
        for (int r = 0; r < 8; ++r) {
            float sq = sqn_m[r] + sq_n - 2.0f * acc[r];
            sq = fmaxf(sq, 0.0f);
            const float dist = sqrtf(sq);
            const int  m    = mBase + r + 8 * hl;
            const bool same = (lbl_m[r] == lbl_n);
            const bool self = (m == n);
            if (same && !self) maxap[r] = fmaxf(maxap[r], dist);
            if (!same)         minan[r] = fminf(minan[r], dist);
        }
    }

    // Reduce across the 16 lanes holding the same row (xor tree stays in half).
#pragma unroll
    for (int r = 0; r < 8; ++r) {
#pragma unroll
        for (int off = 1; off < 16; off <<= 1) {
            maxap[r] = fmaxf(maxap[r], __shfl_xor(maxap[r], off, 32));
            minan[r] = fminf(minan[r], __shfl_xor(minan[r], off, 32));
        }
    }

    // lane 0 -> rows 0..7 of this wave's partials, lane 16 -> rows 8..15
    if (ln == 0) {
#pragma unroll
        for (int r = 0; r < 8; ++r) {
            pAP[wave][hl * 8 + r] = maxap[r];
            pAN[wave][hl * 8 + r] = minan[r];
        }
    }
    __syncthreads();

    // One thread per row: combine 8 waves, apply fallbacks, accumulate loss.
    if (tid < 16) {
        float ap = -__builtin_inff();
        float an =  __builtin_inff();
#pragma unroll
        for (int w = 0; w < 8; ++w) {
            ap = fmaxf(ap, pAP[w][tid]);
            an = fminf(an, pAN[w][tid]);
        }
        const float d_ap = (ap == -__builtin_inff()) ? 0.0f : ap;
        const float d_an = (an ==  __builtin_inff()) ? NEG_FALLBACK_F : an;
        const float row_loss = fmaxf(0.0f, MARGIN_F + d_ap - d_an);
        atomicAdd(out, row_loss * (1.0f / (float)NROWS));
    }
}

// ---------------------------------------------------------------------------
extern "C" void kernel_launch(void* const* d_in, const int* in_sizes, int n_in,
                              void* d_out, int out_size, void* d_ws, size_t ws_size,
                              hipStream_t stream) {
    const float* feat   = (const float*)d_in[0];
    const int*   labels = (const int*)d_in[1];
    float*       out    = (float*)d_out;
    float*       sqn    = (float*)d_ws;     // NROWS floats = 32 KB scratch

    sqnorm_kernel<<<NROWS / 8, 256, 0, stream>>>(feat, sqn, out);
    triplet_hard_kernel<<<NROWS / 16, 256, 0, stream>>>(feat, labels, sqn, out);
}